// AttentionBlock_4853313045194
// MI455X (gfx1250) — compile-verified
//
#include <hip/hip_runtime.h>
#include <hip/hip_bf16.h>
#include <math.h>

#define BB 8
#define LL 2048
#define DD 512
#define SK 40
#define NTOP 40
#define NTOPP 48   // padded to 3 WMMA row-tiles

typedef __attribute__((ext_vector_type(2))) float v2f;
typedef __attribute__((ext_vector_type(8))) float v8f;

static __device__ __forceinline__ v8f wmma4(v2f a, v2f b, v8f c) {
  // D = A(16x4,f32) * B(4x16,f32) + C(16x16,f32)
  return __builtin_amdgcn_wmma_f32_16x16x4_f32(false, a, false, b, (short)0, c,
                                               false, false);
}

// ---------------------------------------------------------------------------
// Kernel 1: fused QKV projection.  Q = x*Wq^T etc.
// One wave -> 16x32 tile of Q, K and V (A fragment shared across 6 WMMAs).
// grid: (B*L/16) mtiles * (DD/32) ntiles = 1024*16 waves = 2048 blocks x 256.
// ---------------------------------------------------------------------------
__global__ __launch_bounds__(256)
void qkv_proj_kernel(const float* __restrict__ x,
                     const float* __restrict__ Wq,
                     const float* __restrict__ Wk,
                     const float* __restrict__ Wv,
                     float* __restrict__ Q, float* __restrict__ K,
                     float* __restrict__ V) {
  const int w    = blockIdx.x * 8 + (threadIdx.x >> 5);
  const int lane = threadIdx.x & 31;
  const int mt = w >> 4, nt = w & 15;
  const int m0 = mt * 16, n0 = nt * 32;
  const int ml = lane & 15, kh = lane >> 4;

  const float* xr  = x  + (size_t)(m0 + ml) * DD + 2 * kh;
  const float* q0  = Wq + (size_t)(n0 + ml) * DD + 2 * kh;
  const float* q1  = Wq + (size_t)(n0 + 16 + ml) * DD + 2 * kh;
  const float* k0p = Wk + (size_t)(n0 + ml) * DD + 2 * kh;
  const float* k1p = Wk + (size_t)(n0 + 16 + ml) * DD + 2 * kh;
  const float* v0p = Wv + (size_t)(n0 + ml) * DD + 2 * kh;
  const float* v1p = Wv + (size_t)(n0 + 16 + ml) * DD + 2 * kh;

  v8f cq0 = {}, cq1 = {}, ck0 = {}, ck1 = {}, cv0 = {}, cv1 = {};
#pragma unroll 4
  for (int k = 0; k < DD; k += 4) {
    v2f a = *(const v2f*)(xr + k);
    cq0 = wmma4(a, *(const v2f*)(q0 + k),  cq0);
    cq1 = wmma4(a, *(const v2f*)(q1 + k),  cq1);
    ck0 = wmma4(a, *(const v2f*)(k0p + k), ck0);
    ck1 = wmma4(a, *(const v2f*)(k1p + k), ck1);
    cv0 = wmma4(a, *(const v2f*)(v0p + k), cv0);
    cv1 = wmma4(a, *(const v2f*)(v1p + k), cv1);
  }
#pragma unroll
  for (int v = 0; v < 8; ++v) {
    const int row = m0 + v + 8 * kh;
    const size_t base = (size_t)row * DD + n0 + ml;
    Q[base]      = cq0[v];  Q[base + 16] = cq1[v];
    K[base]      = ck0[v];  K[base + 16] = ck1[v];
    V[base]      = cv0[v];  V[base + 16] = cv1[v];
  }
}

// ---------------------------------------------------------------------------
// Kernel 2: M[b,l] = max_s(Q[b,l].K[b,idx[l,s]]) - sum_s(...)/L
// One wave per (b,l).  Q row cached in registers (16 f32/lane).
// ---------------------------------------------------------------------------
__global__ __launch_bounds__(256)
void sparse_m_kernel(const float* __restrict__ Q, const float* __restrict__ K,
                     const int* __restrict__ idxs, float* __restrict__ Ms) {
  const int w    = blockIdx.x * 8 + (threadIdx.x >> 5);
  const int lane = threadIdx.x & 31;
  const int b = w >> 11, l = w & (LL - 1);

  const float* qr = Q + (size_t)(b * LL + l) * DD;
  float q[16];
#pragma unroll
  for (int j = 0; j < 16; ++j) q[j] = qr[j * 32 + lane];

  float maxv = -INFINITY, sumv = 0.f;
  for (int s = 0; s < SK; ++s) {
    const int ki = idxs[l * SK + s];
    const float* kr = K + (size_t)(b * LL + ki) * DD;
    float p = 0.f;
#pragma unroll
    for (int j = 0; j < 16; ++j) p += q[j] * kr[j * 32 + lane];
#pragma unroll
    for (int off = 16; off; off >>= 1) p += __shfl_xor(p, off, 32);
    maxv = fmaxf(maxv, p);
    sumv += p;
  }
  if (lane == 0) Ms[b * LL + l] = maxv - sumv * (1.0f / LL);
}

// ---------------------------------------------------------------------------
// Kernel 3: top-NTOP per batch, value-descending, lower index wins ties
// (matches jax.lax.top_k ordering).  One block per batch.
// ---------------------------------------------------------------------------
__global__ __launch_bounds__(256)
void topk_kernel(const float* __restrict__ Ms, int* __restrict__ mtop) {
  __shared__ float sv[LL];
  __shared__ float rv[256];
  __shared__ int   ri[256];
  const int b = blockIdx.x, t = threadIdx.x;
  for (int i = t; i < LL; i += 256) sv[i] = Ms[b * LL + i];
  __syncthreads();
  for (int s = 0; s < NTOP; ++s) {
    float bv = -INFINITY; int bi = LL;
    for (int i = t; i < LL; i += 256) {
      const float v = sv[i];
      if (v > bv || (v == bv && i < bi)) { bv = v; bi = i; }
    }
    rv[t] = bv; ri[t] = bi;
    __syncthreads();
    for (int off = 128; off > 0; off >>= 1) {
      if (t < off) {
        if (rv[t + off] > rv[t] ||
            (rv[t + off] == rv[t] && ri[t + off] < ri[t])) {
          rv[t] = rv[t + off]; ri[t] = ri[t + off];
        }
      }
      __syncthreads();
    }
    if (t == 0) { mtop[b * NTOP + s] = ri[0]; sv[ri[0]] = -INFINITY; }
    __syncthreads();
  }
}

// ---------------------------------------------------------------------------
// Kernel 4a/4b: mean of V over L (two-stage), 4c: broadcast-fill output.
// ---------------------------------------------------------------------------
__global__ __launch_bounds__(256)
void vmean_partial_kernel(const float* __restrict__ V, float* __restrict__ part) {
  const int blk = blockIdx.x;          // B*16 blocks, 128 rows each
  const int b = blk >> 4, c = blk & 15;
  const int l0 = c * 128;
#pragma unroll
  for (int dd = 0; dd < 2; ++dd) {
    const int d = threadIdx.x + dd * 256;
    float s = 0.f;
    for (int l = 0; l < 128; ++l) s += V[(size_t)(b * LL + l0 + l) * DD + d];
    part[(size_t)(b * 16 + c) * DD + d] = s;
  }
}

__global__ __launch_bounds__(256)
void vmean_final_kernel(const float* __restrict__ part, float* __restrict__ vmean) {
  const int b = blockIdx.x;
#pragma unroll
  for (int dd = 0; dd < 2; ++dd) {
    const int d = threadIdx.x + dd * 256;
    float s = 0.f;
#pragma unroll
    for (int c = 0; c < 16; ++c) s += part[(size_t)(b * 16 + c) * DD + d];
    vmean[b * DD + d] = s * (1.0f / LL);
  }
}

__global__ __launch_bounds__(256)
void fill_ctx_kernel(const float* __restrict__ vmean, float* __restrict__ out) {
  const size_t i = (size_t)blockIdx.x * 256 + threadIdx.x;  // B*L*D elems
  const int d = (int)(i & (DD - 1));
  const int b = (int)(i / ((size_t)LL * DD));
  out[i] = vmean[b * DD + d];
}

// ---------------------------------------------------------------------------
// Kernel 5: scores = (1/sqrt(D)) * Q[mtop] . K^T   (per batch: 48 x 2048,
// K-dim 512).  Wave -> 16x32 tile.  waves = B*3*64 = 1536 -> 192 blocks.
// ---------------------------------------------------------------------------
__global__ __launch_bounds__(256)
void scores_kernel(const float* __restrict__ Q, const float* __restrict__ K,
                   const int* __restrict__ mtop, float* __restrict__ sc) {
  const int w    = blockIdx.x * 8 + (threadIdx.x >> 5);
  const int lane = threadIdx.x & 31;
  const int b = w / 192, r = w % 192;
  const int mt = r >> 6, nt = r & 63;
  const int m0 = mt * 16, n0 = nt * 32;
  const int ml = lane & 15, kh = lane >> 4;

  const int u = m0 + ml;
  const int qrow = (u < NTOP) ? mtop[b * NTOP + u] : 0;
  const float* ar = Q + (size_t)(b * LL + qrow) * DD + 2 * kh;
  const float* b0 = K + (size_t)(b * LL + n0 + ml) * DD + 2 * kh;
  const float* b1 = K + (size_t)(b * LL + n0 + 16 + ml) * DD + 2 * kh;

  v8f c0 = {}, c1 = {};
#pragma unroll 4
  for (int k = 0; k < DD; k += 4) {
    v2f a = *(const v2f*)(ar + k);
    c0 = wmma4(a, *(const v2f*)(b0 + k), c0);
    c1 = wmma4(a, *(const v2f*)(b1 + k), c1);
  }
  const float scale = 1.0f / sqrtf((float)DD);
#pragma unroll
  for (int v = 0; v < 8; ++v) {
    const int row = m0 + v + 8 * kh;
    const size_t base = ((size_t)b * NTOPP + row) * LL + n0 + ml;
    sc[base]      = c0[v] * scale;
    sc[base + 16] = c1[v] * scale;
  }
}

// ---------------------------------------------------------------------------
// Kernel 6: softmax over the 2048 keys, one wave per (b,u), u < NTOP.
// ---------------------------------------------------------------------------
__global__ __launch_bounds__(256)
void softmax_kernel(float* __restrict__ sc) {
  const int w    = blockIdx.x * 8 + (threadIdx.x >> 5);  // 0..B*NTOP-1
  const int lane = threadIdx.x & 31;
  const int b = w / NTOP, u = w % NTOP;
  float* row = sc + ((size_t)b * NTOPP + u) * LL;

  float m = -INFINITY;
  for (int j = lane; j < LL; j += 32) m = fmaxf(m, row[j]);
#pragma unroll
  for (int off = 16; off; off >>= 1) m = fmaxf(m, __shfl_xor(m, off, 32));

  float s = 0.f;
  for (int j = lane; j < LL; j += 32) {
    const float e = __expf(row[j] - m);
    row[j] = e;
    s += e;
  }
#pragma unroll
  for (int off = 16; off; off >>= 1) s += __shfl_xor(s, off, 32);
  const float inv = 1.0f / s;
  for (int j = lane; j < LL; j += 32) row[j] *= inv;
}

// ---------------------------------------------------------------------------
// Kernel 7: upd = attn @ V, scattered into out rows mtop.
// Per batch M=48, N=512, K=2048.  waves = B*3*16 = 384 -> 48 blocks.
// ---------------------------------------------------------------------------
__global__ __launch_bounds__(256)
void upd_kernel(const float* __restrict__ att, const float* __restrict__ V,
                const int* __restrict__ mtop, float* __restrict__ out) {
  const int w    = blockIdx.x * 8 + (threadIdx.x >> 5);
  const int lane = threadIdx.x & 31;
  const int b = w / 48, r = w % 48;
  const int mt = r >> 4, nt = r & 15;
  const int m0 = mt * 16, n0 = nt * 32;
  const int ml = lane & 15, kh = lane >> 4;

  const float* ar = att + ((size_t)b * NTOPP + m0 + ml) * LL + 2 * kh;
  v8f c0 = {}, c1 = {};
#pragma unroll 2
  for (int k = 0; k < LL; k += 4) {
    v2f a = *(const v2f*)(ar + k);
    const size_t kr = (size_t)(b * LL + k + 2 * kh);
    v2f bv0, bv1;
    bv0.x = V[kr * DD + n0 + ml];
    bv0.y = V[(kr + 1) * DD + n0 + ml];
    bv1.x = V[kr * DD + n0 + 16 + ml];
    bv1.y = V[(kr + 1) * DD + n0 + 16 + ml];
    c0 = wmma4(a, bv0, c0);
    c1 = wmma4(a, bv1, c1);
  }
#pragma unroll
  for (int v = 0; v < 8; ++v) {
    const int u = m0 + v + 8 * kh;
    if (u < NTOP) {
      const int orow = mtop[b * NTOP + u];
      const size_t base = (size_t)(b * LL + orow) * DD + n0 + ml;
      out[base]      = c0[v];
      out[base + 16] = c1[v];
    }
  }
}

// ---------------------------------------------------------------------------
extern "C" void kernel_launch(void* const* d_in, const int* in_sizes, int n_in,
                              void* d_out, int out_size, void* d_ws,
                              size_t ws_size, hipStream_t stream) {
  const float* x   = (const float*)d_in[0];
  const float* Wq  = (const float*)d_in[1];
  const float* Wk  = (const float*)d_in[2];
  const float* Wv  = (const float*)d_in[3];
  const int*   idx = (const int*)d_in[4];
  float* out = (float*)d_out;

  // workspace carving (floats)
  const size_t QL = (size_t)BB * LL * DD;             // 8388608
  float* ws    = (float*)d_ws;
  float* Qb    = ws;
  float* Kb    = ws + QL;
  float* Vb    = ws + 2 * QL;
  float* Ms    = ws + 3 * QL;                         // B*L
  float* part  = Ms + (size_t)BB * LL;                // B*16*D
  float* vmean = part + (size_t)BB * 16 * DD;         // B*D
  int*   mtop  = (int*)(vmean + (size_t)BB * DD);     // B*NTOP (pad 512)
  float* sc    = (float*)((int*)mtop + 512);          // B*NTOPP*L

  // 1) fused QKV projections (WMMA f32 16x16x4)
  qkv_proj_kernel<<<2048, 256, 0, stream>>>(x, Wq, Wk, Wv, Qb, Kb, Vb);
  // 2) sampled sparsity measure M
  sparse_m_kernel<<<2048, 256, 0, stream>>>(Qb, Kb, idx, Ms);
  // 3) per-batch top-k
  topk_kernel<<<BB, 256, 0, stream>>>(Ms, mtop);
  // 4) V mean + broadcast fill of context
  vmean_partial_kernel<<<BB * 16, 256, 0, stream>>>(Vb, part);
  vmean_final_kernel<<<BB, 256, 0, stream>>>(part, vmean);
  fill_ctx_kernel<<<(unsigned)(QL / 256), 256, 0, stream>>>(vmean, out);
  // 5) scores GEMM (WMMA), 6) softmax, 7) attn@V GEMM + scatter (WMMA)
  scores_kernel<<<192, 256, 0, stream>>>(Qb, Kb, mtop, sc);
  softmax_kernel<<<BB * NTOP / 8, 256, 0, stream>>>(sc);
  upd_kernel<<<48, 256, 0, stream>>>(sc, Vb, mtop, out);
}